// TexualEmbeddingLayer_18399639896074
// MI455X (gfx1250) — compile-verified
//
#include <hip/hip_runtime.h>
#include <hip/hip_bf16.h>

typedef float v2f __attribute__((ext_vector_type(2)));
typedef float v8f __attribute__((ext_vector_type(8)));

// async global->LDS builtin operand types (per hipcc diagnostic: v4i pointers)
typedef int v4i __attribute__((vector_size(4 * sizeof(int))));
typedef __attribute__((address_space(1))) v4i* g_v4ip;
typedef __attribute__((address_space(3))) v4i* l_v4ip;

#define LSEQ 512
#define BS   128
#define DIM  512
#define EDIM 1024
#define HDIM 512
#define KSEL 153
#define MROWS (BS * KSEL)      // 19584 = 306 * 64
#define MBLK  (MROWS / 64)     // 306
#define NROWS_F 19584.0f
#define LDSS 20                // padded LDS row stride (floats)

#if defined(__AMDGCN__) && __has_builtin(__builtin_amdgcn_global_load_async_to_lds_b128)
#define HAVE_ASYNC_LDS 1
#else
#define HAVE_ASYNC_LDS 0
#endif

// 16B global -> LDS copy: async (ASYNCcnt-tracked direct-to-LDS path) when available.
__device__ __forceinline__ void cp16(const float* g, float* l) {
#if HAVE_ASYNC_LDS
  __builtin_amdgcn_global_load_async_to_lds_b128((g_v4ip)g, (l_v4ip)l, 0, 0);
#else
  *(float4*)l = *(const float4*)g;
#endif
}

__device__ __forceinline__ void async_wait_all() {
#if HAVE_ASYNC_LDS
  asm volatile("s_wait_asynccnt 0x0" ::: "memory");
#endif
}

// ---------------------------------------------------------------------------
// Kernel 1: eos argmax + masked score row + rank-based top-k(153)
// ---------------------------------------------------------------------------
__global__ __launch_bounds__(512) void k_topk(const int* __restrict__ text,
                                              const float* __restrict__ atten,
                                              int* __restrict__ idx_out) {
  __shared__ int   sval[LSEQ];
  __shared__ int   sidx[LSEQ];
  __shared__ float sc[LSEQ];
  __shared__ int   s_eos;
  const int b = blockIdx.x;
  const int t = threadIdx.x;

  const int tv = text[b * LSEQ + t];
  sval[t] = tv;
  sidx[t] = t;
  __syncthreads();
  // argmax, first-occurrence semantics (tie -> smaller index)
  for (int s = 256; s > 0; s >>= 1) {
    if (t < s) {
      int v1 = sval[t + s], i1 = sidx[t + s];
      int v0 = sval[t],     i0 = sidx[t];
      if (v1 > v0 || (v1 == v0 && i1 < i0)) { sval[t] = v1; sidx[t] = i1; }
    }
    __syncthreads();
  }
  if (t == 0) s_eos = sidx[0];
  __syncthreads();
  const int eos = s_eos;

  float rowv = (t == eos) ? -1.0f
                          : atten[((size_t)b * LSEQ + eos) * LSEQ + t];
  float m = (tv != 0) ? 1.0f : 0.0f;
  sc[t] = rowv * m;
  __syncthreads();

  // rank = #{i: s_i > s_j} + #{i<j: s_i == s_j}  -> matches lax.top_k order
  const float sj = sc[t];
  int rank = 0;
  for (int i = 0; i < LSEQ; ++i) {
    float si = sc[i];
    rank += (si > sj || (si == sj && i < t)) ? 1 : 0;
  }
  if (rank < KSEL) idx_out[b * KSEL + rank] = t;
}

// ---------------------------------------------------------------------------
// Kernel 2: gather selected rows + L2 normalize (one block per output row)
// ---------------------------------------------------------------------------
__global__ __launch_bounds__(128) void k_gather_norm(const float* __restrict__ features,
                                                     const int* __restrict__ idx,
                                                     float* __restrict__ feats) {
  __shared__ float red[128];
  const int row = blockIdx.x;
  const int b   = row / KSEL;
  const int src = idx[row];
  const int t   = threadIdx.x;

  const float4* fsrc = (const float4*)(features + ((size_t)b * LSEQ + src) * DIM);
  float4 v = fsrc[t];
  red[t] = v.x * v.x + v.y * v.y + v.z * v.z + v.w * v.w;
  __syncthreads();
  for (int s = 64; s > 0; s >>= 1) {
    if (t < s) red[t] += red[t + s];
    __syncthreads();
  }
  float inv = 1.0f / (sqrtf(red[0]) + 1e-8f);
  float4 o;
  o.x = v.x * inv; o.y = v.y * inv; o.z = v.z * inv; o.w = v.w * inv;
  ((float4*)(feats + (size_t)row * DIM))[t] = o;
}

// ---------------------------------------------------------------------------
// fp32 WMMA helper: D = A(16x4) * B(4x16) + C  via v_wmma_f32_16x16x4_f32
// ---------------------------------------------------------------------------
__device__ __forceinline__ v8f wmma4(v2f a, v2f b, v8f c) {
  return __builtin_amdgcn_wmma_f32_16x16x4_f32(false, a, false, b,
                                               (short)0, c, false, false);
}

// ---------------------------------------------------------------------------
// Kernel 3: X0 = feats @ W0^T + b0   (M=19584, N=512, K=512)
// 128 threads = 4 waves; block tile 64x64; each wave: 2x2 tiles of 16x16.
// Double-buffered LDS, async global->LDS prefetch of stage s+1 overlapped
// with WMMA compute of stage s.
// ---------------------------------------------------------------------------
__global__ __launch_bounds__(128) void k_gemm1(const float* __restrict__ A,
                                               const float* __restrict__ Bw,
                                               const float* __restrict__ bias,
                                               float* __restrict__ Cout) {
  __shared__ float lA[2][64 * LDSS];
  __shared__ float lB[2][64 * LDSS];
  const int K = DIM, NOUT = HDIM, NSTAGE = DIM / 16;
  const int n0 = blockIdx.x * 64;
  const int m0 = blockIdx.y * 64;
  const int t = threadIdx.x;
  const int lrow = t >> 1, lhalf = t & 1;
  const int lane = t & 31, wave = t >> 5;
  const int wm = (wave >> 1) * 32, wn = (wave & 1) * 32;
  const int lr = lane & 15, kh = (lane >> 4) * 2, h2 = lane >> 4;

  const float* gAr = A  + (size_t)(m0 + lrow) * K + lhalf * 8;
  const float* gBr = Bw + (size_t)(n0 + lrow) * K + lhalf * 8;
  const int soff = lrow * LDSS + lhalf * 8;

  v8f acc[2][2] = {};

  // prefetch stage 0
  cp16(gAr, &lA[0][soff]);     cp16(gAr + 4, &lA[0][soff] + 4);
  cp16(gBr, &lB[0][soff]);     cp16(gBr + 4, &lB[0][soff] + 4);

  for (int s = 0; s < NSTAGE; ++s) {
    async_wait_all();
    __syncthreads();
    if (s + 1 < NSTAGE) {
      const int nb = (s + 1) & 1;
      const int ko = (s + 1) * 16;
      cp16(gAr + ko, &lA[nb][soff]);  cp16(gAr + ko + 4, &lA[nb][soff] + 4);
      cp16(gBr + ko, &lB[nb][soff]);  cp16(gBr + ko + 4, &lB[nb][soff] + 4);
    }
    const float* bufA = lA[s & 1];
    const float* bufB = lB[s & 1];
#pragma unroll
    for (int kk = 0; kk < 4; ++kk) {
      int kl = kk * 4 + kh;
      v2f aF0, aF1, bF0, bF1;
      aF0.x = bufA[(wm + lr) * LDSS + kl];      aF0.y = bufA[(wm + lr) * LDSS + kl + 1];
      aF1.x = bufA[(wm + 16 + lr) * LDSS + kl]; aF1.y = bufA[(wm + 16 + lr) * LDSS + kl + 1];
      bF0.x = bufB[(wn + lr) * LDSS + kl];      bF0.y = bufB[(wn + lr) * LDSS + kl + 1];
      bF1.x = bufB[(wn + 16 + lr) * LDSS + kl]; bF1.y = bufB[(wn + 16 + lr) * LDSS + kl + 1];
      acc[0][0] = wmma4(aF0, bF0, acc[0][0]);
      acc[0][1] = wmma4(aF0, bF1, acc[0][1]);
      acc[1][0] = wmma4(aF1, bF0, acc[1][0]);
      acc[1][1] = wmma4(aF1, bF1, acc[1][1]);
    }
  }
#pragma unroll
  for (int ti = 0; ti < 2; ++ti)
#pragma unroll
    for (int tj = 0; tj < 2; ++tj) {
      int col = n0 + wn + tj * 16 + lr;
      float bb = bias[col];
#pragma unroll
      for (int v = 0; v < 8; ++v) {
        int row = m0 + wm + ti * 16 + h2 * 8 + v;
        Cout[(size_t)row * NOUT + col] = acc[ti][tj][v] + bb;
      }
    }
}

// ---------------------------------------------------------------------------
// Kernel 4a/4b: deterministic two-stage BN statistics -> per-col affine (a,c)
// ---------------------------------------------------------------------------
__global__ __launch_bounds__(256) void k_colstat_partial(const float* __restrict__ x0,
                                                         float* __restrict__ part) {
  const int blk = blockIdx.x;     // 306 blocks of 64 rows
  const int t   = threadIdx.x;    // cols t and t+256
  const int r0  = blk * 64;
  float s0 = 0.f, q0 = 0.f, s1 = 0.f, q1 = 0.f;
  for (int r = 0; r < 64; ++r) {
    const float* rowp = x0 + (size_t)(r0 + r) * HDIM;
    float v0 = rowp[t], v1 = rowp[t + 256];
    s0 += v0; q0 += v0 * v0;
    s1 += v1; q1 += v1 * v1;
  }
  float* p = part + (size_t)blk * HDIM * 2;
  p[t * 2] = s0;          p[t * 2 + 1] = q0;
  p[(t + 256) * 2] = s1;  p[(t + 256) * 2 + 1] = q1;
}

__global__ __launch_bounds__(512) void k_colstat_final(const float* __restrict__ part,
                                                       const float* __restrict__ g0,
                                                       const float* __restrict__ be0,
                                                       float* __restrict__ aArr,
                                                       float* __restrict__ cArr) {
  const int col = threadIdx.x;
  float s = 0.f, q = 0.f;
  for (int blk = 0; blk < MBLK; ++blk) {
    s += part[(size_t)blk * HDIM * 2 + col * 2];
    q += part[(size_t)blk * HDIM * 2 + col * 2 + 1];
  }
  float mean = s / NROWS_F;
  float var  = fmaxf(q / NROWS_F - mean * mean, 0.0f);
  float a = g0[col] / sqrtf(var + 1e-5f);
  float c = be0[col] - mean * a;
  aArr[col] = a;
  cArr[col] = c;
}

// ---------------------------------------------------------------------------
// Kernel 5: out = feats@W_lin^T + relu(X0*a+c)@W1^T + (b_lin + b1)
// 64 pipelined stages: stages 0..31 accumulate feats@Wlin, 32..63 accumulate
// relu(X0*a+c)@W1 (BN+ReLU applied to A fragments after the LDS load, using
// LDS-cached coefficients). Same async double-buffered pipeline as k_gemm1.
// ---------------------------------------------------------------------------
__global__ __launch_bounds__(128) void k_gemm2(const float* __restrict__ feats,
                                               const float* __restrict__ Wlin,
                                               const float* __restrict__ x0,
                                               const float* __restrict__ W1,
                                               const float* __restrict__ aArr,
                                               const float* __restrict__ cArr,
                                               const float* __restrict__ blin,
                                               const float* __restrict__ b1,
                                               float* __restrict__ out) {
  __shared__ float lA[2][64 * LDSS];
  __shared__ float lB[2][64 * LDSS];
  __shared__ float sBNa[HDIM];
  __shared__ float sBNc[HDIM];
  const int K = DIM, NOUT = EDIM, NSTAGE = 2 * (DIM / 16);  // 64
  const int n0 = blockIdx.x * 64;
  const int m0 = blockIdx.y * 64;
  const int t = threadIdx.x;
  const int lrow = t >> 1, lhalf = t & 1;
  const int lane = t & 31, wave = t >> 5;
  const int wm = (wave >> 1) * 32, wn = (wave & 1) * 32;
  const int lr = lane & 15, kh = (lane >> 4) * 2, h2 = lane >> 4;

  // cache BN affine coefficients in LDS (covered by first barrier)
#pragma unroll
  for (int i = 0; i < HDIM / 128; ++i) {
    sBNa[t + i * 128] = aArr[t + i * 128];
    sBNc[t + i * 128] = cArr[t + i * 128];
  }

  const size_t rowA = (size_t)(m0 + lrow) * K + lhalf * 8;
  const size_t rowB = (size_t)(n0 + lrow) * K + lhalf * 8;
  const int soff = lrow * LDSS + lhalf * 8;

  v8f acc[2][2] = {};

  // prefetch stage 0 (phase 0: feats / Wlin)
  cp16(feats + rowA, &lA[0][soff]);  cp16(feats + rowA + 4, &lA[0][soff] + 4);
  cp16(Wlin + rowB, &lB[0][soff]);   cp16(Wlin + rowB + 4, &lB[0][soff] + 4);

  for (int s = 0; s < NSTAGE; ++s) {
    async_wait_all();
    __syncthreads();
    if (s + 1 < NSTAGE) {
      const int sn = s + 1;
      const int nb = sn & 1;
      const int ko = (sn & 31) * 16;
      const float* An = (sn < 32) ? feats : x0;
      const float* Bn = (sn < 32) ? Wlin : W1;
      cp16(An + rowA + ko, &lA[nb][soff]);  cp16(An + rowA + ko + 4, &lA[nb][soff] + 4);
      cp16(Bn + rowB + ko, &lB[nb][soff]);  cp16(Bn + rowB + ko + 4, &lB[nb][soff] + 4);
    }
    const float* bufA = lA[s & 1];
    const float* bufB = lB[s & 1];
    const int k0g = (s & 31) * 16;
#pragma unroll
    for (int kk = 0; kk < 4; ++kk) {
      int kl = kk * 4 + kh;
      v2f aF0, aF1, bF0, bF1;
      aF0.x = bufA[(wm + lr) * LDSS + kl];      aF0.y = bufA[(wm + lr) * LDSS + kl + 1];
      aF1.x = bufA[(wm + 16 + lr) * LDSS + kl]; aF1.y = bufA[(wm + 16 + lr) * LDSS + kl + 1];
      if (s >= 32) {  // uniform branch: EXEC stays all-ones for WMMA
        int kg = k0g + kl;
        v2f aa = *(const v2f*)&sBNa[kg];
        v2f cc = *(const v2f*)&sBNc[kg];
        aF0.x = fmaxf(0.0f, fmaf(aF0.x, aa.x, cc.x));
        aF0.y = fmaxf(0.0f, fmaf(aF0.y, aa.y, cc.y));
        aF1.x = fmaxf(0.0f, fmaf(aF1.x, aa.x, cc.x));
        aF1.y = fmaxf(0.0f, fmaf(aF1.y, aa.y, cc.y));
      }
      bF0.x = bufB[(wn + lr) * LDSS + kl];      bF0.y = bufB[(wn + lr) * LDSS + kl + 1];
      bF1.x = bufB[(wn + 16 + lr) * LDSS + kl]; bF1.y = bufB[(wn + 16 + lr) * LDSS + kl + 1];
      acc[0][0] = wmma4(aF0, bF0, acc[0][0]);
      acc[0][1] = wmma4(aF0, bF1, acc[0][1]);
      acc[1][0] = wmma4(aF1, bF0, acc[1][0]);
      acc[1][1] = wmma4(aF1, bF1, acc[1][1]);
    }
  }
#pragma unroll
  for (int ti = 0; ti < 2; ++ti)
#pragma unroll
    for (int tj = 0; tj < 2; ++tj) {
      int col = n0 + wn + tj * 16 + lr;
      float bb = blin[col] + b1[col];
#pragma unroll
      for (int v = 0; v < 8; ++v) {
        int row = m0 + wm + ti * 16 + h2 * 8 + v;
        out[(size_t)row * NOUT + col] = acc[ti][tj][v] + bb;
      }
    }
}

// ---------------------------------------------------------------------------
extern "C" void kernel_launch(void* const* d_in, const int* in_sizes, int n_in,
                              void* d_out, int out_size, void* d_ws, size_t ws_size,
                              hipStream_t stream) {
  const float* features = (const float*)d_in[0];
  const int*   text     = (const int*)d_in[1];
  const float* atten    = (const float*)d_in[2];
  const float* W_lin    = (const float*)d_in[3];
  const float* b_lin    = (const float*)d_in[4];
  const float* W0       = (const float*)d_in[5];
  const float* b0       = (const float*)d_in[6];
  const float* g0       = (const float*)d_in[7];
  const float* be0      = (const float*)d_in[8];
  const float* W1       = (const float*)d_in[9];
  const float* b1       = (const float*)d_in[10];
  float* out = (float*)d_out;

  // workspace layout
  char* ws = (char*)d_ws;
  const size_t FEAT_BYTES = (size_t)MROWS * DIM * sizeof(float);   // 40,108,032
  const size_t PART_BYTES = (size_t)MBLK * HDIM * 2 * sizeof(float);
  int*   idx   = (int*)ws;                                         // 19584 ints
  float* feats = (float*)(ws + (1 << 17));
  float* x0    = (float*)(ws + (1 << 17) + FEAT_BYTES);
  float* part  = (float*)(ws + (1 << 17) + 2 * FEAT_BYTES);
  float* aArr  = (float*)(ws + (1 << 17) + 2 * FEAT_BYTES + PART_BYTES);
  float* cArr  = aArr + HDIM;

  k_topk<<<BS, 512, 0, stream>>>(text, atten, idx);
  k_gather_norm<<<MROWS, 128, 0, stream>>>(features, idx, feats);
  dim3 g1(HDIM / 64, MBLK);
  k_gemm1<<<g1, 128, 0, stream>>>(feats, W0, b0, x0);
  k_colstat_partial<<<MBLK, 256, 0, stream>>>(x0, part);
  k_colstat_final<<<1, 512, 0, stream>>>(part, g0, be0, aArr, cArr);
  dim3 g2(EDIM / 64, MBLK);
  k_gemm2<<<g2, 128, 0, stream>>>(feats, W_lin, x0, W1, aArr, cArr, b_lin, b1, out);
}